// StyleGAN2_3D_Generator_70806830842188
// MI455X (gfx1250) — compile-verified
//
#include <hip/hip_runtime.h>
#include <math.h>

typedef __attribute__((ext_vector_type(2))) float v2f;
typedef __attribute__((ext_vector_type(4))) float v4f;
typedef __attribute__((ext_vector_type(8))) float v8f;

#define NB   4
#define PP   32768
#define NN   (NB * PP)        // 131072 points
#define CIN  128
#define COUT 128
#define WD   512
#define K3   27
#define LDA  132              // padded LDS row stride (floats), 8B-aligned, bank-spread
#define MT   128              // rows per block

// ---------------- styles -> sn (modulation vector) ----------------
__global__ void styles_kernel(const float* __restrict__ w,
                              const float* __restrict__ aw,
                              const float* __restrict__ ab,
                              float* __restrict__ sn) {
  __shared__ float red[128];
  const int c = threadIdx.x;                       // channel 0..127
  const float scale = 0.044194173824159216f;       // 1/sqrt(512)
  float st[NB];
  for (int b = 0; b < NB; ++b) {
    float acc = 0.f;
    const float* wp = w + b * WD;
    const float* ap = aw + c * WD;
    for (int d = 0; d < WD; ++d) acc += wp[d] * ap[d];
    st[b] = acc * scale + ab[c];
  }
  float ss = 0.f;
  for (int b = 0; b < NB; ++b) ss += st[b] * st[b];
  red[c] = ss;
  __syncthreads();
  for (int s = 64; s > 0; s >>= 1) {
    if (c < s) red[c] += red[c + s];
    __syncthreads();
  }
  const float r = rsqrtf(red[0] / (float)(NB * CIN));
  for (int b = 0; b < NB; ++b) sn[b * CIN + c] = st[b] * r;
}

// ------------- weight normalization + transpose to [k][cin][cout] -------------
__global__ void wnorm_kernel(const float* __restrict__ cw,
                             float* __restrict__ wkT,
                             float* __restrict__ s2) {
  __shared__ float red[128];
  const int co = blockIdx.x;
  const int ci = threadIdx.x;
  float wv[K3];
  const float* p = cw + ((size_t)co * CIN + ci) * K3;
  float ss = 0.f;
  for (int k = 0; k < K3; ++k) { wv[k] = p[k]; ss += wv[k] * wv[k]; }
  red[ci] = ss;
  __syncthreads();
  for (int s = 64; s > 0; s >>= 1) {
    if (ci < s) red[ci] += red[ci + s];
    __syncthreads();
  }
  const float r = rsqrtf(red[0] / (float)(CIN * K3));
  float s2a = 0.f;
  for (int k = 0; k < K3; ++k) {
    const float wn = wv[k] * r;
    wkT[((size_t)k * CIN + ci) * COUT + co] = wn;
    s2a += wn * wn;
  }
  s2[co * CIN + ci] = s2a;
}

// ---------------- demodulation coefficients ----------------
__global__ void dcoef_kernel(const float* __restrict__ sn,
                             const float* __restrict__ s2,
                             float* __restrict__ dc) {
  const int t  = threadIdx.x;       // 0..511
  const int b  = t >> 7;
  const int co = t & 127;
  float acc = 0.f;
  for (int ci = 0; ci < CIN; ++ci) {
    const float s = sn[b * CIN + ci];
    acc += s * s * s2[co * CIN + ci];
  }
  dc[b * COUT + co] = rsqrtf(acc + 1e-8f);
}

// ------------- gather -> modulate -> GEMM (WMMA f32) -> scatter, one tap -------------
// Block: 256 threads = 8 waves = 2 M-groups(64 rows) x 4 N-panels(32 cols).
// Each wave: 4 M-subtiles x 2 col-tiles = 8 independent WMMA accumulation chains.
__global__ __launch_bounds__(256)
void spconv_pass_kernel(const float* __restrict__ x,
                        const float* __restrict__ sn,
                        const float* __restrict__ wkT,
                        const int* __restrict__ in_idx,
                        const int* __restrict__ out_idx,
                        float* __restrict__ out,
                        float* __restrict__ dummy,
                        int kpass) {
  const int i0 = blockIdx.x * MT;
  const int* __restrict__ inI  = in_idx  + (size_t)kpass * NN;
  const int* __restrict__ outI = out_idx + (size_t)kpass * NN;

  // valid entries are a prefix of each tap's list: one scalar test skips the tile
  if (outI[i0] >= NN) return;

  __shared__ float sA[MT * LDA];
  __shared__ int   sOut[MT];

  const int t = threadIdx.x;
  if (t < MT) sOut[t] = outI[i0 + t];

  // Stage 128 gathered rows x 128ch into LDS, applying per-batch modulation.
  {
    const int rgrp = t >> 5;           // 0..7
    const int c4   = (t & 31) * 4;     // channel group
    for (int j = 0; j < 16; ++j) {
      const int row = j * 8 + rgrp;
      const int g = inI[i0 + row];
      v4f val = {0.f, 0.f, 0.f, 0.f};
      if (g < NN) {
        const float* xp = x + (size_t)g * CIN + c4;
        const float* sp = sn + ((g >> 15) * CIN) + c4;   // batch = row >> 15
        val.x = xp[0] * sp[0];
        val.y = xp[1] * sp[1];
        val.z = xp[2] * sp[2];
        val.w = xp[3] * sp[3];
      }
      *(v4f*)&sA[row * LDA + c4] = val;
    }
  }
  __syncthreads();

  const int wave  = t >> 5;
  const int mg    = wave >> 2;           // M-group: 0 or 1 (64 rows each)
  const int wp    = wave & 3;            // N-panel: 32 cols each
  const int lane  = t & 31;
  const int half  = lane >> 4;
  const int rbase = mg * 64;
  const int nc    = wp * 32 + (lane & 15);   // first col-tile; second is nc+16

  // Seed accumulators from the output buffer (read-modify-write per tap).
  v8f acc[8];
  for (int sub = 0; sub < 4; ++sub) {
    for (int v = 0; v < 8; ++v) {
      const int rl   = rbase + sub * 16 + v + 8 * half;
      const int orow = sOut[rl];
      const size_t roff = ((size_t)orow) << 7;          // computed unconditionally
      const float* p0 = out + roff + nc;
      const float* p1 = dummy + nc;
      const float* cp = (orow < NN) ? p0 : p1;          // pointer cndmask
      acc[sub * 2 + 0][v] = cp[0];
      acc[sub * 2 + 1][v] = cp[16];
    }
  }

  const float* __restrict__ Bbase = wkT + (size_t)kpass * CIN * COUT;

  #pragma unroll 4
  for (int kk = 0; kk < 32; ++kk) {
    const int k0 = kk * 4 + half * 2;
    // B fragments: B[k0][nc], B[k0+1][nc] from [k][cin][cout] layout (L2-hot)
    const float* bp = Bbase + (size_t)k0 * COUT + nc;
    v2f b0, b1;
    b0.x = bp[0];
    b0.y = bp[COUT];
    b1.x = bp[16];
    b1.y = bp[COUT + 16];
    #pragma unroll
    for (int sub = 0; sub < 4; ++sub) {
      const int m = rbase + sub * 16 + (lane & 15);
      const v2f afrag = *(const v2f*)&sA[m * LDA + k0];
      acc[sub * 2 + 0] = __builtin_amdgcn_wmma_f32_16x16x4_f32(
          false, afrag, false, b0, (short)0, acc[sub * 2 + 0], false, false);
      acc[sub * 2 + 1] = __builtin_amdgcn_wmma_f32_16x16x4_f32(
          false, afrag, false, b1, (short)0, acc[sub * 2 + 1], false, false);
    }
  }

  for (int sub = 0; sub < 4; ++sub) {
    for (int v = 0; v < 8; ++v) {
      const int rl   = rbase + sub * 16 + v + 8 * half;
      const int orow = sOut[rl];
      const size_t roff = ((size_t)orow) << 7;
      float* p0 = out + roff + nc;
      float* p1 = dummy + nc;
      float* cp = (orow < NN) ? p0 : p1;
      cp[0]  = acc[sub * 2 + 0][v];
      cp[16] = acc[sub * 2 + 1][v];
    }
  }
}

// ---------------- epilogue: demod, bias, leaky-relu*sqrt2, clip (in place) ----------------
__global__ void epilogue_kernel(float* __restrict__ out,
                                const float* __restrict__ dc,
                                const float* __restrict__ bias,
                                const float* __restrict__ mag) {
  const size_t idx = (size_t)blockIdx.x * blockDim.x + threadIdx.x;
  if (idx >= (size_t)NN * COUT) return;
  const int c = (int)(idx & 127);
  const int i = (int)(idx >> 7);
  const int b = i >> 15;
  float v = out[idx];
  v = v * dc[b * COUT + c] * rsqrtf(mag[0]) + bias[c];
  v = (v < 0.f ? 0.2f * v : v) * 1.4142135623730951f;
  v = fminf(fmaxf(v, -256.f), 256.f);
  out[idx] = v;
}

extern "C" void kernel_launch(void* const* d_in, const int* in_sizes, int n_in,
                              void* d_out, int out_size, void* d_ws, size_t ws_size,
                              hipStream_t stream) {
  const float* x_feat = (const float*)d_in[0];
  const float* w      = (const float*)d_in[1];
  const float* aw     = (const float*)d_in[2];
  const float* ab     = (const float*)d_in[3];
  const float* cw     = (const float*)d_in[4];
  const float* cbias  = (const float*)d_in[5];
  const float* mag    = (const float*)d_in[6];
  const int*   in_idx  = (const int*)d_in[8];
  const int*   out_idx = (const int*)d_in[9];
  float* out = (float*)d_out;

  // workspace carve-up (floats)
  float* ws    = (float*)d_ws;
  float* wkT   = ws;                                   // 27*128*128
  float* sn    = wkT + (size_t)K3 * CIN * COUT;        // 4*128
  float* s2    = sn + NB * CIN;                        // 128*128
  float* dc    = s2 + COUT * CIN;                      // 4*128
  float* dummy = dc + NB * COUT;                       // 128 (sink row)

  hipMemsetAsync(d_out, 0, (size_t)NN * COUT * sizeof(float), stream);

  styles_kernel<<<1, 128, 0, stream>>>(w, aw, ab, sn);
  wnorm_kernel<<<COUT, 128, 0, stream>>>(cw, wkT, s2);
  dcoef_kernel<<<1, 512, 0, stream>>>(sn, s2, dc);

  for (int k = 0; k < K3; ++k) {
    spconv_pass_kernel<<<NN / MT, 256, 0, stream>>>(
        x_feat, sn, wkT, in_idx, out_idx, out, dummy, k);
  }

  const size_t total = (size_t)NN * COUT;
  epilogue_kernel<<<(unsigned)((total + 255) / 256), 256, 0, stream>>>(
      out, dc, cbias, mag);
}